// TemporalGraphEuler_89352499626124
// MI455X (gfx1250) — compile-verified
//
#include <hip/hip_runtime.h>
#include <hip/hip_bf16.h>
#include <math.h>

#define N_NODES 8192
#define D_H     256
#define K_NN    10

typedef __attribute__((ext_vector_type(2))) float v2f;
typedef __attribute__((ext_vector_type(8))) float v8f;

// ---------------------------------------------------------------------------
// KNN: one block per query node. Distances cached in LDS, then 10 rounds of
// block-wide argmin (strict < => lowest index wins ties, like lax.top_k).
// Uses the reference formula d = |pi|^2 + |pj|^2 - 2*dot.
// ---------------------------------------------------------------------------
__global__ __launch_bounds__(128)
void knn_kernel(const float* __restrict__ pos, int* __restrict__ nbr)
{
    __shared__ float sdist[N_NODES];
    __shared__ float rv[128];
    __shared__ int   ri[128];
    const int i   = blockIdx.x;
    const int tid = threadIdx.x;
    const float xi = pos[i*3+0], yi = pos[i*3+1], zi = pos[i*3+2];
    const float sqi = xi*xi + yi*yi + zi*zi;
    for (int j = tid; j < N_NODES; j += 128) {
        const float xj = pos[j*3+0], yj = pos[j*3+1], zj = pos[j*3+2];
        const float sqj = xj*xj + yj*yj + zj*zj;
        const float dot = xi*xj + yi*yj + zi*zj;
        sdist[j] = (j == i) ? INFINITY : (sqi + sqj - 2.0f*dot);
    }
    __syncthreads();
    for (int t = 0; t < K_NN; ++t) {
        float best = INFINITY; int bidx = 0;
        const int j0 = tid * (N_NODES/128);
        for (int j = j0; j < j0 + N_NODES/128; ++j) {
            const float d = sdist[j];
            if (d < best) { best = d; bidx = j; }
        }
        rv[tid] = best; ri[tid] = bidx;
        __syncthreads();
        for (int s = 64; s > 0; s >>= 1) {
            if (tid < s) {
                if (rv[tid+s] < rv[tid]) { rv[tid] = rv[tid+s]; ri[tid] = ri[tid+s]; }
            }
            __syncthreads();
        }
        if (tid == 0) {
            nbr[i*K_NN + t] = ri[0];
            sdist[ri[0]] = INFINITY;   // remove winner for next round
        }
        __syncthreads();
    }
}

// ---------------------------------------------------------------------------
// Embedding: H = landmarks @ emb_w + emb_b   (D_IN = 3)
// ---------------------------------------------------------------------------
__global__ __launch_bounds__(256)
void emb_kernel(const float* __restrict__ L, const float* __restrict__ Ew,
                const float* __restrict__ Eb, float* __restrict__ H)
{
    const int gid = blockIdx.x * 256 + threadIdx.x;   // N_NODES*D_H threads
    const int i = gid >> 8, j = gid & 255;
    float v = fmaf(L[i*3+2], Ew[2*D_H+j],
              fmaf(L[i*3+1], Ew[1*D_H+j],
              fmaf(L[i*3+0], Ew[0*D_H+j], Eb[j])));
    H[gid] = v;
}

// ---------------------------------------------------------------------------
// Neighbor mean: Xout[i,:] = (1/10) * sum_t Xin[nbr[i][t],:]
// (deg == 10 for every node, so gcn_norm == 1/10 on every edge)
// ---------------------------------------------------------------------------
__global__ __launch_bounds__(256)
void agg_kernel(const float* __restrict__ Xin, const int* __restrict__ nbr,
                float* __restrict__ Xout)
{
    const int gid = blockIdx.x * 256 + threadIdx.x;   // N_NODES*64 threads
    const int i = gid >> 6;
    const int c = (gid & 63) << 2;
    float4 acc = make_float4(0.f, 0.f, 0.f, 0.f);
    #pragma unroll
    for (int t = 0; t < K_NN; ++t) {
        const int n = nbr[i*K_NN + t];
        const float4 v = *(const float4*)(Xin + (size_t)n*D_H + c);
        acc.x += v.x; acc.y += v.y; acc.z += v.z; acc.w += v.w;
    }
    const float nrm = 0.1f;
    *(float4*)(Xout + (size_t)i*D_H + c) =
        make_float4(acc.x*nrm, acc.y*nrm, acc.z*nrm, acc.w*nrm);
}

// ---------------------------------------------------------------------------
// Fused TAGConv GEMM + Euler update, fp32 WMMA 16x16x4.
//   C = H@W0 + X1@W1 + X2@W2 + conv_b ;  Hout = H + 0.1*tanh(C)
// Block: 8 waves, one N-tile (16 cols); W slices staged in 48KB LDS.
// Each wave owns TWO 16x16 M-tiles (B-fragment reuse).
// A frag (16x4 f32): lane gives M=lane&15, float2 at K = k0 + 2*(lane>>4).
// B frag (4x16 f32): VGPR0/1 hold rows K/K+1, N = lane&15, K offset by lane half.
// C/D: elem r -> (M = 8*(lane>>4)+r, N = lane&15).
// ---------------------------------------------------------------------------
__global__ __launch_bounds__(256)
void gemm_update_kernel(const float* __restrict__ H,
                        const float* __restrict__ X1,
                        const float* __restrict__ X2,
                        const float* __restrict__ convW,   // [3][256][256]
                        const float* __restrict__ convB,   // [256]
                        float* __restrict__ Hout)
{
    __shared__ float Wlds[3][D_H][16];
    const int tid = threadIdx.x;
    const int n0  = blockIdx.y * 16;
    #pragma unroll
    for (int m = 0; m < 3; ++m)
        for (int idx = tid; idx < D_H*16; idx += 256) {
            const int k = idx >> 4, n = idx & 15;
            Wlds[m][k][n] = convW[m*D_H*D_H + k*D_H + n0 + n];
        }
    __syncthreads();

    const int wave = tid >> 5;
    const int lane = tid & 31;
    const int mt0  = (blockIdx.x * 8 + wave) * 2;     // first of two M-tiles
    const int mrow = lane & 15;
    const int kgrp = (lane >> 4) << 1;                // 0 or 2
    const int ncol = lane & 15;

    const float bias = convB[n0 + ncol];
    v8f acc0 = { bias, bias, bias, bias, bias, bias, bias, bias };
    v8f acc1 = acc0;

    const float* Xs[3] = { H, X1, X2 };
    const size_t r0 = (size_t)(mt0*16 + mrow) * D_H + kgrp;
    const size_t r1 = r0 + (size_t)16 * D_H;

    for (int m = 0; m < 3; ++m) {
        const float* __restrict__ X = Xs[m];
        #pragma unroll 4
        for (int k0 = 0; k0 < D_H; k0 += 4) {
            const int kb = k0 + kgrp;
            v2f b;
            b.x = Wlds[m][kb][ncol];
            b.y = Wlds[m][kb+1][ncol];
            const v2f a0 = *(const v2f*)(X + r0 + k0);
            const v2f a1 = *(const v2f*)(X + r1 + k0);
            acc0 = __builtin_amdgcn_wmma_f32_16x16x4_f32(
                false, a0, false, b, (short)0, acc0, false, false);
            acc1 = __builtin_amdgcn_wmma_f32_16x16x4_f32(
                false, a1, false, b, (short)0, acc1, false, false);
        }
    }

    const int mbase = mt0*16 + ((lane >> 4) << 3);
    #pragma unroll
    for (int r = 0; r < 8; ++r) {
        const size_t o0 = (size_t)(mbase + r) * D_H + n0 + ncol;
        const size_t o1 = o0 + (size_t)16 * D_H;
        Hout[o0] = H[o0] + 0.1f * tanhf(acc0[r]);
        Hout[o1] = H[o1] + 0.1f * tanhf(acc1[r]);
    }
}

// ---------------------------------------------------------------------------
// Readout: out[i,:] = H[i,:] @ W(256x3) + b.  One wave32 per row.
// ---------------------------------------------------------------------------
__global__ __launch_bounds__(256)
void readout_kernel(const float* __restrict__ H, const float* __restrict__ W,
                    const float* __restrict__ b, float* __restrict__ out)
{
    const int row  = (blockIdx.x * 256 + threadIdx.x) >> 5;
    const int lane = threadIdx.x & 31;
    if (row >= N_NODES) return;
    float a0 = 0.f, a1 = 0.f, a2 = 0.f;
    for (int k = lane; k < D_H; k += 32) {
        const float h = H[(size_t)row*D_H + k];
        a0 = fmaf(h, W[k*3+0], a0);
        a1 = fmaf(h, W[k*3+1], a1);
        a2 = fmaf(h, W[k*3+2], a2);
    }
    #pragma unroll
    for (int off = 16; off > 0; off >>= 1) {
        a0 += __shfl_xor(a0, off, 32);
        a1 += __shfl_xor(a1, off, 32);
        a2 += __shfl_xor(a2, off, 32);
    }
    if (lane == 0) {
        out[row*3+0] = a0 + b[0];
        out[row*3+1] = a1 + b[1];
        out[row*3+2] = a2 + b[2];
    }
}

__global__ __launch_bounds__(256)
void copy_kernel(const float* __restrict__ src, float* __restrict__ dst)
{
    const int gid = blockIdx.x * 256 + threadIdx.x;
    dst[gid] = src[gid];
}

// ---------------------------------------------------------------------------
extern "C" void kernel_launch(void* const* d_in, const int* in_sizes, int n_in,
                              void* d_out, int out_size, void* d_ws, size_t ws_size,
                              hipStream_t stream)
{
    const float* landmarks = (const float*)d_in[0];   // [8192,3]
    const float* emb_w     = (const float*)d_in[1];   // [3,256]
    const float* emb_b     = (const float*)d_in[2];   // [256]
    const float* readout_w = (const float*)d_in[3];   // [256,3]
    const float* readout_b = (const float*)d_in[4];   // [3]
    const float* conv_ws   = (const float*)d_in[5];   // [3,256,256]
    const float* conv_b    = (const float*)d_in[6];   // [256]

    // Output layout: y [8192*3] | h_final [8192*256] | x [8192*3] | land_pred [10*8192*3]
    float* out     = (float*)d_out;
    float* y_out   = out;
    float* hf_out  = y_out + N_NODES*3;
    float* x_out   = hf_out + N_NODES*D_H;
    float* lp_out  = x_out + N_NODES*3;

    // Workspace: nbr | H_a | H_b | X1 | X2   (~34 MB)
    char*  ws  = (char*)d_ws;
    int*   nbr = (int*)ws;
    float* Ha  = (float*)(ws + (((size_t)N_NODES*K_NN*sizeof(int) + 255) & ~(size_t)255));
    float* Hb  = Ha + (size_t)N_NODES*D_H;
    float* X1  = Hb + (size_t)N_NODES*D_H;
    float* X2  = X1 + (size_t)N_NODES*D_H;

    knn_kernel<<<N_NODES, 128, 0, stream>>>(landmarks, nbr);
    emb_kernel<<<N_NODES, 256, 0, stream>>>(landmarks, emb_w, emb_b, Ha);
    readout_kernel<<<1024, 256, 0, stream>>>(Ha, readout_w, readout_b, x_out);

    float* cur = Ha;
    float* nxt = Hb;
    for (int t = 0; t < 10; ++t) {
        agg_kernel<<<2048, 256, 0, stream>>>(cur, nbr, X1);
        agg_kernel<<<2048, 256, 0, stream>>>(X1, nbr, X2);
        gemm_update_kernel<<<dim3(32, 16), 256, 0, stream>>>(
            cur, X1, X2, conv_ws, conv_b, nxt);
        readout_kernel<<<1024, 256, 0, stream>>>(
            nxt, readout_w, readout_b, lp_out + (size_t)t*N_NODES*3);
        float* tmp = cur; cur = nxt; nxt = tmp;
    }
    readout_kernel<<<1024, 256, 0, stream>>>(cur, readout_w, readout_b, y_out);
    copy_kernel<<<N_NODES, 256, 0, stream>>>(cur, hf_out);
}